// MultiheadAttnBlock_35493609734835
// MI455X (gfx1250) — compile-verified
//
#include <hip/hip_runtime.h>
#include <hip/hip_bf16.h>

// ---------------------------------------------------------------------------
// MultiheadAttnBlock for MI455X (gfx1250), fp32 end-to-end via
// V_WMMA_F32_16X16X4_F32 (exact-precision fp32 matrix core path).
//
// Shapes (fixed by the reference): B=8, C=512, H=W=32 -> S=1024,
// heads=8, hd=64, 3C=1536, GN groups=32 (16 ch/group).
//
// WMMA 16x16x4 f32 register layouts (per CDNA5 ISA 7.12.2, codegen-verified):
//   A (16x4):  lane L holds row m=L&15; k-base kb=2*(L>>4); vgpr j -> K=kb+j
//   B (4x16):  lane L holds col n=L&15; k-base kb=2*(L>>4); vgpr j -> K=kb+j
//   C/D(16x16):vgpr r, lane L -> element [m = r + 8*(L>>4)][n = L&15]
// ---------------------------------------------------------------------------

typedef __attribute__((ext_vector_type(2))) float v2f;
typedef __attribute__((ext_vector_type(8))) float v8f;

__device__ __forceinline__ v8f wmma_f32_4(v2f a, v2f b, v8f c) {
  // (neg_a, A, neg_b, B, c_mod, C, reuse_a, reuse_b)
  return __builtin_amdgcn_wmma_f32_16x16x4_f32(false, a, false, b, (short)0, c,
                                               false, false);
}

#define B_DIM 8
#define C_DIM 512
#define S_DIM 1024
#define NH    8
#define HD    64
#define D3    1536

// ---------------------------------------------------------------------------
// Kernel 1: GroupNorm. One block per (batch, group). 16 ch * 1024 px = 16384.
// ---------------------------------------------------------------------------
__global__ void __launch_bounds__(256)
gn_kernel(const float* __restrict__ x, const float* __restrict__ scale,
          const float* __restrict__ bias, float* __restrict__ out) {
  const int tid = threadIdx.x;
  const int b = blockIdx.x >> 5;
  const int g = blockIdx.x & 31;
  const size_t base = ((size_t)b * C_DIM + g * 16) * S_DIM;

  __shared__ float ssum[256];
  __shared__ float ssq[256];

  float s = 0.f, q = 0.f;
  for (int i = tid; i < 16 * S_DIM; i += 256) {
    float v = x[base + i];
    s += v;
    q += v * v;
  }
  ssum[tid] = s;
  ssq[tid] = q;
  __syncthreads();
  for (int off = 128; off > 0; off >>= 1) {
    if (tid < off) {
      ssum[tid] += ssum[tid + off];
      ssq[tid] += ssq[tid + off];
    }
    __syncthreads();
  }
  const float inv_n = 1.0f / (16.0f * S_DIM);
  const float mean = ssum[0] * inv_n;
  const float var = ssq[0] * inv_n - mean * mean;
  const float rinv = rsqrtf(var + 1e-6f);

  for (int i = tid; i < 16 * S_DIM; i += 256) {
    const int c = g * 16 + (i >> 10);
    const float v = x[base + i];
    out[base + i] = (v - mean) * rinv * scale[c] + bias[c];
  }
}

// ---------------------------------------------------------------------------
// Kernel 2: 1x1-conv GEMM with register blocking.
//   out[b,d,s] = sum_c Wt[c,d]*in[b,c,s] + bias[d]  (+ resid if HAS_RES)
// One wave computes a 64(d) x 32(s) block = 4x2 WMMA tiles.
// Per K-step (K=4): 8 A-loads + 4 B-loads -> 8 WMMAs (1.5 loads/WMMA).
// ---------------------------------------------------------------------------
template <int DOUT, bool HAS_RES>
__global__ void __launch_bounds__(256)
nin_gemm_kernel(const float* __restrict__ in, const float* __restrict__ Wt,
                const float* __restrict__ bias, const float* __restrict__ resid,
                float* __restrict__ out) {
  constexpr int CIN = 512;
  constexpr int MT = 4;                    // m-tiles (d): 64 rows
  constexpr int NT = 2;                    // n-tiles (s): 32 cols
  constexpr int STILES = S_DIM / (16 * NT);   // 32
  constexpr int DTILES = DOUT / (16 * MT);

  const int lane = threadIdx.x & 31;
  const int w = threadIdx.x >> 5;
  int t = blockIdx.x * 8 + w;

  const int s_t = t % STILES;
  t /= STILES;
  const int d_t = t % DTILES;
  const int b = t / DTILES;

  const int s0 = s_t * 16 * NT;
  const int d0 = d_t * 16 * MT;
  const int nl = lane & 15;
  const int hi = lane >> 4;
  const int kb = hi * 2;                   // K base for this half-wave

  const v8f vzero = {0.f, 0.f, 0.f, 0.f, 0.f, 0.f, 0.f, 0.f};
  v8f acc[MT][NT];
#pragma unroll
  for (int mi = 0; mi < MT; ++mi)
#pragma unroll
    for (int ni = 0; ni < NT; ++ni) acc[mi][ni] = vzero;

  const float* Ap = Wt + (size_t)kb * DOUT + d0 + nl;              // A[m,k]
  const float* Bp = in + ((size_t)b * CIN + kb) * S_DIM + s0 + nl; // B[k,n]

  for (int c0 = 0; c0 < CIN; c0 += 4) {
    v2f a[MT];
    v2f bb[NT];
#pragma unroll
    for (int mi = 0; mi < MT; ++mi) {
      a[mi].x = Ap[mi * 16];
      a[mi].y = Ap[mi * 16 + DOUT];
    }
#pragma unroll
    for (int ni = 0; ni < NT; ++ni) {
      bb[ni].x = Bp[ni * 16];
      bb[ni].y = Bp[ni * 16 + S_DIM];
    }
    __builtin_prefetch(Ap + (size_t)8 * DOUT, 0, 3);   // A panel 2 steps ahead
#pragma unroll
    for (int mi = 0; mi < MT; ++mi)
#pragma unroll
      for (int ni = 0; ni < NT; ++ni)
        acc[mi][ni] = wmma_f32_4(a[mi], bb[ni], acc[mi][ni]);
    Ap += (size_t)4 * DOUT;
    Bp += 4 * S_DIM;
  }

  const int hi8 = hi * 8;
#pragma unroll
  for (int mi = 0; mi < MT; ++mi) {
#pragma unroll
    for (int r = 0; r < 8; ++r) {
      const int d = d0 + mi * 16 + r + hi8;
      const float bv = bias[d];
#pragma unroll
      for (int ni = 0; ni < NT; ++ni) {
        const size_t oidx =
            ((size_t)b * DOUT + d) * S_DIM + s0 + ni * 16 + nl;
        float v = acc[mi][ni][r] + bv;
        if (HAS_RES) v += resid[oidx];
        out[oidx] = v;
      }
    }
  }
}

// ---------------------------------------------------------------------------
// Kernel 3: flash attention. One wave per (b, head, 16-query tile).
// qkv layout: (B, 1536, S); head hh -> q ch = hh*192, k = +64, v = +128.
// attn out layout (B, C, S): attn[b, hh*64+d, s].
// ---------------------------------------------------------------------------
__global__ void __launch_bounds__(256)
attn_kernel(const float* __restrict__ qkv, float* __restrict__ attn) {
  __shared__ float lds_p[8][16][17];   // per-wave P tile  [query][key]
  __shared__ float lds_v[8][16][80];   // per-wave V tile  [key][feat]

  const int lane = threadIdx.x & 31;
  const int w = threadIdx.x >> 5;
  int id = blockIdx.x * 8 + w;
  const int qt = id & 63;
  id >>= 6;
  const int hh = id & 7;
  const int b = id >> 3;

  const int q0 = qt * 16;
  const int nl = lane & 15;
  const int hi = lane >> 4;
  const int kb = hi * 2;
  const int hi8 = hi * 8;

  const size_t chq = (size_t)b * D3 + hh * 192;
  const size_t chk = chq + 64;
  const size_t chv = chq + 128;

  // Preload Q A-fragments: A[m = query, k = feature], 16 K-chunks of 4.
  v2f qa[16];
#pragma unroll
  for (int kc = 0; kc < 16; ++kc) {
    const float* qp = qkv + (chq + 4 * kc + kb) * S_DIM + q0 + nl;
    qa[kc].x = qp[0];
    qa[kc].y = qp[S_DIM];
  }

  float rmax[8], rsum[8];
#pragma unroll
  for (int r = 0; r < 8; ++r) {
    rmax[r] = -3.0e38f;
    rsum[r] = 0.f;
  }
  v8f o[4];
  const v8f vzero = {0.f, 0.f, 0.f, 0.f, 0.f, 0.f, 0.f, 0.f};
#pragma unroll
  for (int dt = 0; dt < 4; ++dt) o[dt] = vzero;

  for (int kt = 0; kt < 64; ++kt) {
    const int n0 = kt * 16;

    // Stage V tile [16 keys][64 feats] into wave-private LDS (coalesced in s).
#pragma unroll 4
    for (int i = 0; i < 32; ++i) {
      const int d = 2 * i + hi;
      lds_v[w][nl][d] = qkv[(chv + d) * S_DIM + n0 + nl];
    }

    // S_tile = Q * K^T  (B[k = feature, n = key])
    v8f st = vzero;
#pragma unroll
    for (int kc = 0; kc < 16; ++kc) {
      const float* kp = qkv + (chk + 4 * kc + kb) * S_DIM + n0 + nl;
      v2f bf;
      bf.x = kp[0];
      bf.y = kp[S_DIM];
      st = wmma_f32_4(qa[kc], bf, st);
    }

    // Online softmax: row reductions across the 16-lane half that owns a row.
    float alpha[8];
#pragma unroll
    for (int r = 0; r < 8; ++r) {
      const float v = st[r] * 0.125f;   // 1/sqrt(64)
      float mx = v;
      mx = fmaxf(mx, __shfl_xor(mx, 1, 32));
      mx = fmaxf(mx, __shfl_xor(mx, 2, 32));
      mx = fmaxf(mx, __shfl_xor(mx, 4, 32));
      mx = fmaxf(mx, __shfl_xor(mx, 8, 32));
      const float mnew = fmaxf(rmax[r], mx);
      const float a = __expf(rmax[r] - mnew);
      const float p = __expf(v - mnew);
      rmax[r] = mnew;
      float ps = p;
      ps += __shfl_xor(ps, 1, 32);
      ps += __shfl_xor(ps, 2, 32);
      ps += __shfl_xor(ps, 4, 32);
      ps += __shfl_xor(ps, 8, 32);
      rsum[r] = rsum[r] * a + ps;
      alpha[r] = a;
      lds_p[w][r + hi8][nl] = p;        // D-layout -> [query][key] in LDS
    }
#pragma unroll
    for (int dt = 0; dt < 4; ++dt)
#pragma unroll
      for (int r = 0; r < 8; ++r) o[dt][r] *= alpha[r];

    // O += P * V   (A = P from LDS, B = V from LDS)
#pragma unroll
    for (int kc = 0; kc < 4; ++kc) {
      v2f af;
      af.x = lds_p[w][nl][4 * kc + kb];
      af.y = lds_p[w][nl][4 * kc + kb + 1];
#pragma unroll
      for (int dt = 0; dt < 4; ++dt) {
        v2f bf;
        bf.x = lds_v[w][4 * kc + kb][16 * dt + nl];
        bf.y = lds_v[w][4 * kc + kb + 1][16 * dt + nl];
        o[dt] = wmma_f32_4(af, bf, o[dt]);
      }
    }
  }

  // Write out (B, C, S): attn[b, hh*64 + dt*16 + n, q0 + m]
#pragma unroll
  for (int dt = 0; dt < 4; ++dt) {
#pragma unroll
    for (int r = 0; r < 8; ++r) {
      const size_t ch = (size_t)b * C_DIM + hh * HD + dt * 16 + nl;
      attn[ch * S_DIM + q0 + r + hi8] = o[dt][r] / rsum[r];
    }
  }
}

// ---------------------------------------------------------------------------
// Host launcher. Inputs: x, gn_scale, gn_bias, W_qkv, b_qkv, W_out, b_out,
// num_heads (ignored; fixed 8). Scratch: h_gn 16MB | qkv 48MB | attn 16MB.
// ---------------------------------------------------------------------------
extern "C" void kernel_launch(void* const* d_in, const int* in_sizes, int n_in,
                              void* d_out, int out_size, void* d_ws,
                              size_t ws_size, hipStream_t stream) {
  (void)in_sizes; (void)n_in; (void)out_size; (void)ws_size;

  const float* x        = (const float*)d_in[0];
  const float* gn_scale = (const float*)d_in[1];
  const float* gn_bias  = (const float*)d_in[2];
  const float* W_qkv    = (const float*)d_in[3];
  const float* b_qkv    = (const float*)d_in[4];
  const float* W_out    = (const float*)d_in[5];
  const float* b_out    = (const float*)d_in[6];
  float* out = (float*)d_out;

  float* h_gn = (float*)d_ws;
  float* qkv  = h_gn + (size_t)B_DIM * C_DIM * S_DIM;  // +4M floats
  float* attn = qkv + (size_t)B_DIM * D3 * S_DIM;      // +12M floats

  // 1) GroupNorm: 8 batches * 32 groups
  gn_kernel<<<256, 256, 0, stream>>>(x, gn_scale, gn_bias, h_gn);

  // 2) QKV projection: 8 b * (1536/64) d-blocks * (1024/32) s-blocks
  //    = 6144 waves / 8 per block = 768 blocks
  nin_gemm_kernel<D3, false><<<768, 256, 0, stream>>>(h_gn, W_qkv, b_qkv,
                                                      nullptr, qkv);

  // 3) Attention: 8 * 8 * 64 = 4096 waves / 8 per block
  attn_kernel<<<512, 256, 0, stream>>>(qkv, attn);

  // 4) Output projection + residual: 8 * (512/64) * 32 = 2048 waves
  nin_gemm_kernel<C_DIM, true><<<256, 256, 0, stream>>>(attn, W_out, b_out, x,
                                                        out);
}